// GNN_23407571763695
// MI455X (gfx1250) — compile-verified
//
#include <hip/hip_runtime.h>
#include <math.h>

typedef __attribute__((ext_vector_type(2))) float v2f;
typedef __attribute__((ext_vector_type(8))) float v8f;

#define HFEAT 128
#define BN_EPS 1e-5f

// ---------------------------------------------------------------------------
// Edge scatter:  agg[dst] += h[src]  (agg pre-initialized with h -> A + I)
// one (edge, 4-float chunk) per thread; 32 chunks cover the 128 features.
// ---------------------------------------------------------------------------
__global__ void gnn_edge_scatter(const long long* __restrict__ ei,
                                 const float* __restrict__ h,
                                 float* __restrict__ agg,
                                 long long E) {
  long long t = (long long)blockIdx.x * blockDim.x + threadIdx.x;
  if (t >= E * 32) return;
  long long e = t >> 5;
  int c = (int)(t & 31);
  int dst = (int)ei[e];         // row 0 of edge_index
  int src = (int)ei[E + e];     // row 1 of edge_index
  const float4 v = *(const float4*)(h + (long long)src * HFEAT + c * 4);
  float* p = agg + (long long)dst * HFEAT + c * 4;
  atomicAdd(p + 0, v.x);
  atomicAdd(p + 1, v.y);
  atomicAdd(p + 2, v.z);
  atomicAdd(p + 3, v.w);
}

// ---------------------------------------------------------------------------
// Zero helper (graph-capture safe init of BN accumulators)
// ---------------------------------------------------------------------------
__global__ void gnn_zero(float* __restrict__ p, int n) {
  int t = blockIdx.x * blockDim.x + threadIdx.x;
  if (t < n) p[t] = 0.0f;
}

// ---------------------------------------------------------------------------
// Y = A(32-row stripe) @ W(128x128) + bias via fp32 WMMA 16x16x4.
// Block = 256 threads = 8 waves; wave w owns output columns [16w, 16w+16)
// and TWO 16-row m-tiles -> each B fragment feeds 2 independent WMMAs
// (2-way ILP, half the B-load traffic per FLOP).
// Fused: per-column sum / sum-of-squares atomics for the following BatchNorm.
// ---------------------------------------------------------------------------
__global__ __launch_bounds__(256) void gnn_gemm128(
    const float* __restrict__ A, const float* __restrict__ W,
    const float* __restrict__ bias, float* __restrict__ Y,
    float* __restrict__ colsum, float* __restrict__ colsumsq) {
  __shared__ float As[32 * HFEAT];          // 16 KB stripe of A (32 rows)
  const int m0 = blockIdx.x * 32;
  const int tid = threadIdx.x;

  // cooperative load: 4096 floats = 1024 float4, 4 per thread
  const float4* Ag = (const float4*)(A + (long long)m0 * HFEAT);
  float4* As4 = (float4*)As;
#pragma unroll
  for (int i = 0; i < 4; ++i) As4[tid + 256 * i] = Ag[tid + 256 * i];
  __syncthreads();

  const int lane = tid & 31;
  const int wave = tid >> 5;
  const int l15 = lane & 15;
  const int half = lane >> 4;               // lane group (ISA frag layout)
  const int n = wave * 16 + l15;            // output column of this lane

  const float* Ar0 = As + l15 * HFEAT + 2 * half;      // m-tile 0
  const float* Ar1 = Ar0 + 16 * HFEAT;                 // m-tile 1
  const float* Wc = W + n + 2 * half * HFEAT;          // W[k base][n]

  v8f c0 = {}, c1 = {};
#pragma unroll
  for (int k = 0; k < HFEAT; k += 4) {
    v2f b;
    b.x = Wc[k * HFEAT];
    b.y = Wc[k * HFEAT + HFEAT];
    v2f a0, a1;
    a0.x = Ar0[k];
    a0.y = Ar0[k + 1];
    a1.x = Ar1[k];
    a1.y = Ar1[k + 1];
    c0 = __builtin_amdgcn_wmma_f32_16x16x4_f32(false, a0, false, b,
                                               (short)0, c0, false, false);
    c1 = __builtin_amdgcn_wmma_f32_16x16x4_f32(false, a1, false, b,
                                               (short)0, c1, false, false);
  }

  const float bn = bias[n];
  float s = 0.0f, ss = 0.0f;
  float* Yc0 = Y + (long long)(m0 + 8 * half) * HFEAT + n;
  float* Yc1 = Yc0 + 16 * HFEAT;
#pragma unroll
  for (int v = 0; v < 8; ++v) {             // rows m0 + 8*half + v (+16)
    float v0 = c0[v] + bn;
    float v1 = c1[v] + bn;
    Yc0[(long long)v * HFEAT] = v0;
    Yc1[(long long)v * HFEAT] = v1;
    s += v0 + v1;
    ss += v0 * v0 + v1 * v1;
  }
  atomicAdd(&colsum[n], s);
  atomicAdd(&colsumsq[n], ss);
}

// ---------------------------------------------------------------------------
// BN stats finalize: mean / invstd per column (biased variance, eps=1e-5)
// ---------------------------------------------------------------------------
__global__ void gnn_bn_stats(const float* __restrict__ sum,
                             const float* __restrict__ sumsq,
                             float* __restrict__ mean,
                             float* __restrict__ invstd, float invN) {
  int t = threadIdx.x;
  float m = sum[t] * invN;
  float v = sumsq[t] * invN - m * m;
  mean[t] = m;
  invstd[t] = rsqrtf(v + BN_EPS);
}

// ---------------------------------------------------------------------------
// H = relu((Y - mean) * invstd), float4-vectorized
// ---------------------------------------------------------------------------
__global__ void gnn_bn_relu(const float* __restrict__ Y,
                            const float* __restrict__ mean,
                            const float* __restrict__ invstd,
                            float* __restrict__ H, int nchunks) {
  int t = blockIdx.x * blockDim.x + threadIdx.x;
  if (t >= nchunks) return;
  int c4 = (t & 31) * 4;                    // column of first element
  float4 y = ((const float4*)Y)[t];
  float4 r;
  r.x = fmaxf((y.x - mean[c4 + 0]) * invstd[c4 + 0], 0.0f);
  r.y = fmaxf((y.y - mean[c4 + 1]) * invstd[c4 + 1], 0.0f);
  r.z = fmaxf((y.z - mean[c4 + 2]) * invstd[c4 + 2], 0.0f);
  r.w = fmaxf((y.w - mean[c4 + 3]) * invstd[c4 + 3], 0.0f);
  ((float4*)H)[t] = r;
}

// ---------------------------------------------------------------------------
// Final linear 128 -> OUTF (=40): block = 192 threads = 6 waves
// = 2 m-tiles x 3 n-tiles (cols padded to 48, masked edges).
// EXEC stays all-1s through WMMA (mask only memory ops).
// ---------------------------------------------------------------------------
__global__ __launch_bounds__(192) void gnn_gemm_out(
    const float* __restrict__ A, const float* __restrict__ W,
    const float* __restrict__ bias, float* __restrict__ out, int OUTF) {
  __shared__ float As[32 * HFEAT];          // 16 KB
  const int m0 = blockIdx.x * 32;
  const int tid = threadIdx.x;
  const float4* Ag = (const float4*)(A + (long long)m0 * HFEAT);
  float4* As4 = (float4*)As;
  for (int i = tid; i < 1024; i += 192) As4[i] = Ag[i];
  __syncthreads();

  const int lane = tid & 31;
  const int wave = tid >> 5;                // 0..5
  const int nt = wave % 3;                  // n-tile
  const int mt = wave / 3;                  // m-tile
  const int l15 = lane & 15;
  const int half = lane >> 4;
  const int n = nt * 16 + l15;
  const bool valid = n < OUTF;

  const float* Ar = As + (mt * 16 + l15) * HFEAT + 2 * half;
  const float* Wc = W + (valid ? n : 0) + 2 * half * OUTF;

  v8f c = {};
#pragma unroll
  for (int k = 0; k < HFEAT; k += 4) {
    v2f a, b;
    a.x = Ar[k];
    a.y = Ar[k + 1];
    float w0 = Wc[k * OUTF];
    float w1 = Wc[k * OUTF + OUTF];
    b.x = valid ? w0 : 0.0f;
    b.y = valid ? w1 : 0.0f;
    c = __builtin_amdgcn_wmma_f32_16x16x4_f32(false, a, false, b,
                                              (short)0, c, false, false);
  }

  if (valid) {
    const float bn = bias[n];
    float* Oc = out + (long long)(m0 + mt * 16 + 8 * half) * OUTF + n;
#pragma unroll
    for (int v = 0; v < 8; ++v) Oc[(long long)v * OUTF] = c[v] + bn;
  }
}

// ---------------------------------------------------------------------------
// Orchestration
// ---------------------------------------------------------------------------
extern "C" void kernel_launch(void* const* d_in, const int* in_sizes, int n_in,
                              void* d_out, int out_size, void* d_ws, size_t ws_size,
                              hipStream_t stream) {
  const float* x = (const float*)d_in[0];
  const long long* ei = (const long long*)d_in[1];
  const float* Wm[3] = {(const float*)d_in[2], (const float*)d_in[4],
                        (const float*)d_in[6]};
  const float* Bm[3] = {(const float*)d_in[3], (const float*)d_in[5],
                        (const float*)d_in[7]};
  const float* Wl = (const float*)d_in[8];
  const float* bl = (const float*)d_in[9];
  float* out = (float*)d_out;

  const int N = in_sizes[0] / HFEAT;        // 100000
  const long long E = in_sizes[1] / 2;      // 800000
  const int OUTF = in_sizes[9];             // 40

  const size_t nh = (size_t)N * HFEAT;
  float* bufA = (float*)d_ws;               // aggregated features
  float* bufB = bufA + nh;                  // post-linear
  float* bufC = bufB + nh;                  // post-BN/ReLU (layer output)
  float* sum = bufC + nh;                   // [128]
  float* sumsq = sum + HFEAT;               // [128]
  float* mean = sumsq + HFEAT;              // [128]
  float* invstd = mean + HFEAT;             // [128]

  const int scat_blocks = (int)((E * 32 + 255) / 256);
  const int bn_chunks = N * (HFEAT / 4);
  const int bn_blocks = (bn_chunks + 255) / 256;
  const int row_blocks = N / 32;            // 3125 (N divisible by 32)

  const float* hin = x;
  for (int l = 0; l < 3; ++l) {
    // agg = h (self loop), then scatter-add over edges
    hipMemcpyAsync(bufA, hin, nh * sizeof(float), hipMemcpyDeviceToDevice,
                   stream);
    gnn_edge_scatter<<<scat_blocks, 256, 0, stream>>>(ei, hin, bufA, E);
    // BN accumulators
    gnn_zero<<<1, 256, 0, stream>>>(sum, 2 * HFEAT);
    // linear + bias + fused column stats (WMMA fp32)
    gnn_gemm128<<<row_blocks, 256, 0, stream>>>(bufA, Wm[l], Bm[l], bufB, sum,
                                                sumsq);
    gnn_bn_stats<<<1, HFEAT, 0, stream>>>(sum, sumsq, mean, invstd,
                                          1.0f / (float)N);
    gnn_bn_relu<<<bn_blocks, 256, 0, stream>>>(bufB, mean, invstd, bufC,
                                               bn_chunks);
    hin = bufC;
  }
  gnn_gemm_out<<<row_blocks, 192, 0, stream>>>(bufC, Wl, bl, out, OUTF);
}